// GConv_39685497815679
// MI455X (gfx1250) — compile-verified
//
#include <hip/hip_runtime.h>
#include <math.h>

// ---------------------------------------------------------------------------
// Types for CDNA5 WMMA f32 16x16x4
// ---------------------------------------------------------------------------
typedef __attribute__((ext_vector_type(2))) float v2f;
typedef __attribute__((ext_vector_type(8))) float v8f;

#define LDSP 132  // padded LDS row stride (floats): 132 % 64 = 4 -> conflict-free column reads

// ---------------------------------------------------------------------------
// GEMM: H[N x NCOL] = X[N x 128] @ W[128 x NCOL], fp32 via V_WMMA_F32_16X16X4_F32
// One block per 16-row tile; wave w computes columns [16w, 16w+16).
// ---------------------------------------------------------------------------
template <int NCOL>
__global__ __launch_bounds__((NCOL / 16) * 32)
void gemm_wmma_f32(const float* __restrict__ X, const float* __restrict__ W,
                   float* __restrict__ H, int N) {
  __shared__ float xs[16 * LDSP];
  const int tile = blockIdx.x;
  const int t = threadIdx.x;
  const int nthr = (NCOL / 16) * 32;
  const long long rowbase = (long long)tile * 16;

  // Stage the 16x128 X tile into LDS (coalesced global reads).
  for (int i = t; i < 16 * 128; i += nthr) {
    const int r = i >> 7, c = i & 127;
    const long long gr = rowbase + r;
    xs[r * LDSP + c] = (gr < N) ? X[gr * 128 + c] : 0.0f;
  }
  __syncthreads();

  const int lane = t & 31;
  const int wave = t >> 5;
  const int m = lane & 15;             // A-matrix row within tile
  const int koff = (lane >> 4) * 2;    // lanes 16-31 hold K+2, K+3
  const int col = wave * 16 + (lane & 15);

  v8f acc = {0.f, 0.f, 0.f, 0.f, 0.f, 0.f, 0.f, 0.f};

#pragma unroll
  for (int k = 0; k < 128; k += 4) {
    v2f a, b;
    // A 16x4 fp32 layout: lane 0-15 -> (K=k, K=k+1); lane 16-31 -> (K=k+2, K=k+3)
    a.x = xs[m * LDSP + k + koff];
    a.y = xs[m * LDSP + k + koff + 1];
    // B 4x16 fp32 layout mirrors A along K; N = lane&15 per wave's column slab
    b.x = W[(k + koff) * NCOL + col];
    b.y = W[(k + koff + 1) * NCOL + col];
    acc = __builtin_amdgcn_wmma_f32_16x16x4_f32(false, a, false, b, (short)0,
                                                acc, false, false);
  }

  // C/D layout: VGPR j -> M=j (lanes 0-15) / M=j+8 (lanes 16-31), N = lane&15
  const int msel = (lane >> 4) * 8;
#pragma unroll
  for (int j = 0; j < 8; ++j) {
    const long long row = rowbase + msel + j;
    if (row < N) H[row * NCOL + col] = acc[j];
  }
}

// ---------------------------------------------------------------------------
// Degree / normalization kernels
// ---------------------------------------------------------------------------
__global__ void deg_init(float* __restrict__ deg, int N) {
  int i = blockIdx.x * blockDim.x + threadIdx.x;
  if (i < N) deg[i] = 1.0f;  // self-loop weight
}

__global__ void deg_accum(const int* __restrict__ col, const float* __restrict__ ew,
                          float* __restrict__ deg, int E) {
  const int stride = gridDim.x * blockDim.x;
  for (int e = blockIdx.x * blockDim.x + threadIdx.x; e < E; e += stride)
    atomicAdd(&deg[col[e]], ew[e]);
}

__global__ void deg_rsqrt(const float* __restrict__ deg, float* __restrict__ dinv, int N) {
  int i = blockIdx.x * blockDim.x + threadIdx.x;
  if (i < N) {
    const float d = deg[i];
    dinv[i] = (d > 0.0f) ? rsqrtf(d) : 0.0f;
  }
}

// ---------------------------------------------------------------------------
// out[i,f] = h[i,f] * dinv[i]^2 + bias[f]   (self-loop term + bias)
// ---------------------------------------------------------------------------
template <int LOGF>
__global__ void self_bias_init(const float* __restrict__ h, const float* __restrict__ dinv,
                               const float* __restrict__ bias, float* __restrict__ out,
                               long long total) {
  const long long stride = (long long)gridDim.x * blockDim.x;
  for (long long i = (long long)blockIdx.x * blockDim.x + threadIdx.x; i < total; i += stride) {
    const int f = (int)(i & ((1 << LOGF) - 1));
    const float di = dinv[i >> LOGF];
    out[i] = h[i] * di * di + bias[f];
  }
}

// ---------------------------------------------------------------------------
// Edge scatter: out[c,:] += h[r,:] * (dinv[r]*w*dinv[c]).
// 32 consecutive lanes share one edge (F=128) -> coalesced float4 gather +
// coalesced global_atomic_add_f32 scatter.
// ---------------------------------------------------------------------------
template <int LOGQ>  // quads per row: F = 4 << LOGQ
__global__ void edge_scatter(const float* __restrict__ h, const int* __restrict__ er,
                             const int* __restrict__ ec, const float* __restrict__ ew,
                             const float* __restrict__ dinv, float* __restrict__ out,
                             int E) {
  const int F = 4 << LOGQ;
  const long long total = (long long)E << LOGQ;
  const long long stride = (long long)gridDim.x * blockDim.x;
  for (long long i = (long long)blockIdx.x * blockDim.x + threadIdx.x; i < total; i += stride) {
    const int e = (int)(i >> LOGQ);
    const int q = (int)(i & ((1 << LOGQ) - 1));
    const int r = er[e], c = ec[e];
    const float nrm = dinv[r] * ew[e] * dinv[c];
    const float4 hv = *(const float4*)(h + (long long)r * F + q * 4);
    float* op = out + (long long)c * F + q * 4;
    atomicAdd(op + 0, hv.x * nrm);
    atomicAdd(op + 1, hv.y * nrm);
    atomicAdd(op + 2, hv.z * nrm);
    atomicAdd(op + 3, hv.w * nrm);
  }
}

// ---------------------------------------------------------------------------
// BatchNorm (training stats over nodes) + PReLU
// ---------------------------------------------------------------------------
__global__ void bn_zero(float* __restrict__ stats) {
  stats[threadIdx.x] = 0.0f;  // 256 threads: sum[128], sumsq[128]
}

__global__ void bn_stats(const float* __restrict__ z, float* __restrict__ stats, int N) {
  const int f = threadIdx.x;  // 128 threads
  float s = 0.0f, s2 = 0.0f;
  for (int r = blockIdx.x; r < N; r += gridDim.x) {
    const float v = z[(long long)r * 128 + f];
    s += v;
    s2 += v * v;
  }
  atomicAdd(&stats[f], s);
  atomicAdd(&stats[128 + f], s2);
}

__global__ void bn_finalize(float* __restrict__ stats, const float* __restrict__ gamma,
                            const float* __restrict__ beta, float invN) {
  const int f = threadIdx.x;  // 128 threads
  const float mu = stats[f] * invN;
  const float var = stats[128 + f] * invN - mu * mu;
  const float sc = gamma[f] * rsqrtf(var + 1e-5f);
  stats[256 + f] = sc;              // scale
  stats[384 + f] = beta[f] - mu * sc;  // shift
}

__global__ void bn_prelu_apply(float* __restrict__ z, const float* __restrict__ stats,
                               const float* __restrict__ pa, long long total) {
  const float a = pa[0];
  const float* scale = stats + 256;
  const float* shift = stats + 384;
  const long long stride = (long long)gridDim.x * blockDim.x;
  for (long long i = (long long)blockIdx.x * blockDim.x + threadIdx.x; i < total; i += stride) {
    const int f = (int)(i & 127);
    const float t = z[i] * scale[f] + shift[f];
    z[i] = (t > 0.0f) ? t : a * t;
  }
}

// ---------------------------------------------------------------------------
// Host-side orchestration
// ---------------------------------------------------------------------------
extern "C" void kernel_launch(void* const* d_in, const int* in_sizes, int n_in,
                              void* d_out, int out_size, void* d_ws, size_t ws_size,
                              hipStream_t stream) {
  const float* x     = (const float*)d_in[0];
  const int*   eidx  = (const int*)d_in[1];
  const float* ew    = (const float*)d_in[2];
  const float* W1    = (const float*)d_in[3];
  const float* b1    = (const float*)d_in[4];
  const float* W2    = (const float*)d_in[5];
  const float* b2    = (const float*)d_in[6];
  const float* gamma = (const float*)d_in[7];
  const float* beta  = (const float*)d_in[8];
  const float* pa    = (const float*)d_in[9];

  const int N = in_sizes[0] / 128;  // 50000
  const int E = in_sizes[2];        // 1600000
  const int* er = eidx;             // edge_index[0,:] (source)
  const int* ec = eidx + E;         // edge_index[1,:] (target)

  // Workspace layout (floats)
  float* deg   = (float*)d_ws;
  float* dinv  = deg + N;
  float* h1    = dinv + N;                   // N x 128
  float* z     = h1 + (size_t)N * 128;       // N x 128
  float* h2    = z + (size_t)N * 128;        // N x 64
  float* stats = h2 + (size_t)N * 64;        // 512 floats

  float* out = (float*)d_out;                // N x 64

  const int ntiles = (N + 15) / 16;

  // 1) Symmetric normalization coefficients
  deg_init<<<(N + 255) / 256, 256, 0, stream>>>(deg, N);
  deg_accum<<<2048, 256, 0, stream>>>(ec, ew, deg, E);
  deg_rsqrt<<<(N + 255) / 256, 256, 0, stream>>>(deg, dinv, N);

  // 2) h1 = x @ W1  (fp32 WMMA)
  gemm_wmma_f32<128><<<ntiles, 256, 0, stream>>>(x, W1, h1, N);

  // 3) z = scatter(h1) + b1
  self_bias_init<7><<<2048, 256, 0, stream>>>(h1, dinv, b1, z, (long long)N * 128);
  edge_scatter<5><<<4096, 256, 0, stream>>>(h1, er, ec, ew, dinv, z, E);

  // 4) BatchNorm (batch stats) + PReLU, in place on z
  bn_zero<<<1, 256, 0, stream>>>(stats);
  bn_stats<<<1024, 128, 0, stream>>>(z, stats, N);
  bn_finalize<<<1, 128, 0, stream>>>(stats, gamma, beta, 1.0f / (float)N);
  bn_prelu_apply<<<2048, 256, 0, stream>>>(z, stats, pa, (long long)N * 128);

  // 5) h2 = z @ W2  (fp32 WMMA, 64 output cols)
  gemm_wmma_f32<64><<<ntiles, 128, 0, stream>>>(z, W2, h2, N);

  // 6) out = scatter(h2) + b2
  self_bias_init<6><<<2048, 256, 0, stream>>>(h2, dinv, b2, out, (long long)N * 64);
  edge_scatter<4><<<4096, 256, 0, stream>>>(h2, er, ec, ew, dinv, out, E);
}